// Pool_57174604644521
// MI455X (gfx1250) — compile-verified
//
#include <hip/hip_runtime.h>
#include <hip/hip_bf16.h>
#include <math.h>

// ---------------------------------------------------------------------------
// CDNA5 / gfx1250 implementation of Top-K graph pooling (SAGPool-like).
// Memory-bound workload (~360 MB traffic -> ~15us at 23.3 TB/s). The GEMV
// for the scores uses V_WMMA_F32_16X16X4_F32 (exact fp32 math on the matrix
// pipe); everything else is bandwidth-limited streaming + scans.
// ---------------------------------------------------------------------------

typedef __attribute__((ext_vector_type(2))) float v2f;
typedef __attribute__((ext_vector_type(8))) float v8f;

// -------------------------- score GEMV via WMMA ----------------------------
// D = A(16xK) * B(Kx16) + C, K consumed 4 at a time. B[k][n] = w[k] for every
// column n, so every column of D holds the same dot product; lanes with
// (lane&15)==0 own rows r+8*hi in D VGPRs r=0..7 and write the 16 scores.
//
// A layout (32-bit 16x4): lanes 0-15 hold (K0,K1) of row M=lane,
//                         lanes 16-31 hold (K2,K3) of row M=lane-16.
// B layout (32-bit 4x16): symmetric in K -> per-lane contents depend only on
//                         the same (K0,K1)/(K2,K3) split since B is constant
//                         across columns.
__global__ __launch_bounds__(256) void score_kernel(
    const float* __restrict__ x, const float* __restrict__ w,
    const float* __restrict__ bias, float* __restrict__ scores,
    int N, int F) {
  const int lane = threadIdx.x & 31;
  const int wave = threadIdx.x >> 5;
  const int rowBase = (blockIdx.x * 8 + wave) * 16;
  const int hi = lane >> 4;      // 0: K pair (0,1), 1: K pair (2,3)
  const int m  = lane & 15;

  int row = rowBase + m;
  if (row >= N) row = N - 1;     // clamp so EXEC stays all-ones for WMMA
  const float* xr = x + (size_t)row * F;

  v8f c = {0.f, 0.f, 0.f, 0.f, 0.f, 0.f, 0.f, 0.f};

#if __has_builtin(__builtin_amdgcn_wmma_f32_16x16x4_f32)
  const int nChunks = F >> 2;    // K=4 per WMMA
  for (int t = 0; t < nChunks; ++t) {
    const int k = 4 * t + 2 * hi;
    v2f a; a.x = xr[k];  a.y = xr[k + 1];
    v2f b; b.x = w[k];   b.y = w[k + 1];
    // 8 args: (neg_a, A, neg_b, B, c_mod, C, reuse_a, reuse_b)
    c = __builtin_amdgcn_wmma_f32_16x16x4_f32(false, a, false, b,
                                              (short)0, c, false, false);
  }
  const float b0 = bias[0];
  if (m == 0) {
#pragma unroll
    for (int r = 0; r < 8; ++r) {
      const int orow = rowBase + r + 8 * hi;   // D VGPR r: row r (lo), r+8 (hi)
      if (orow < N) {
        const float s = c[r] + b0;
        scores[orow] = 1.0f / (1.0f + expf(-s));
      }
    }
  }
#else
  // Scalar fallback (should not be taken on gfx1250 toolchains).
  const int orow = rowBase + m + 16 * hi;      // not used; keep simple per-thread
  int rr = blockIdx.x * 256 + threadIdx.x;
  if (rr < N) {
    const float* xp = x + (size_t)rr * F;
    float acc = 0.f;
    for (int k = 0; k < F; ++k) acc += xp[k] * w[k];
    acc += bias[0];
    scores[rr] = 1.0f / (1.0f + expf(-acc));
  }
  (void)orow; (void)c;
#endif
}

// --------------------- order-preserving float -> uint key ------------------
__device__ __forceinline__ unsigned fkey(float f) {
  unsigned b = __float_as_uint(f);
  return (b & 0x80000000u) ? ~b : (b | 0x80000000u);
}

// ----------------------------- radix select --------------------------------
// state[0]=prefix, state[1]=k_remaining (1-based), state[2]=threshold key
__global__ void radix_init_kernel(unsigned* state, unsigned* hist, unsigned k) {
  hist[threadIdx.x] = 0u;
  if (threadIdx.x == 0) { state[0] = 0u; state[1] = k; state[2] = 0u; }
}

__global__ __launch_bounds__(256) void radix_hist_kernel(
    const float* __restrict__ scores, int N,
    const unsigned* __restrict__ state, unsigned* __restrict__ hist, int shift) {
  __shared__ unsigned lh[256];
  lh[threadIdx.x] = 0u;
  __syncthreads();
  const unsigned prefix = state[0];
  const unsigned himask = (shift == 24) ? 0u : (0xFFFFFFFFu << (shift + 8));
  const int i = blockIdx.x * 256 + threadIdx.x;
  if (i < N) {
    const unsigned key = fkey(scores[i]);
    if ((key & himask) == (prefix & himask))
      atomicAdd(&lh[(key >> shift) & 0xFFu], 1u);
  }
  __syncthreads();
  if (lh[threadIdx.x]) atomicAdd(&hist[threadIdx.x], lh[threadIdx.x]);
}

__global__ __launch_bounds__(256) void radix_pick_kernel(
    unsigned* state, unsigned* hist, int shift) {
  __shared__ unsigned sc[256];
  const unsigned kr = state[1];          // read before any write (1 block)
  const unsigned v = hist[threadIdx.x];
  sc[threadIdx.x] = v;
  __syncthreads();
  for (int off = 1; off < 256; off <<= 1) {
    unsigned t = (threadIdx.x >= (unsigned)off) ? sc[threadIdx.x - off] : 0u;
    __syncthreads();
    sc[threadIdx.x] += t;
    __syncthreads();
  }
  const unsigned incl = sc[threadIdx.x];
  const unsigned excl = incl - v;
  if (incl >= kr && excl < kr) {         // exactly one digit wins
    const unsigned d = threadIdx.x;
    const unsigned prefix = state[0] | (d << shift);
    state[0] = prefix;
    state[1] = kr - excl;
    if (shift == 0) state[2] = prefix;   // final 32-bit threshold key
  }
  hist[threadIdx.x] = 0u;                // ready for next pass
}

// --------------------- generic single-block exclusive scan -----------------
__global__ __launch_bounds__(1024) void scan_exclusive_kernel(
    const unsigned* __restrict__ in, unsigned* __restrict__ out, int n) {
  __shared__ unsigned tile[1024];
  __shared__ unsigned carry;
  if (threadIdx.x == 0) carry = 0u;
  __syncthreads();
  for (int base = 0; base < n; base += 1024) {
    const int i = base + (int)threadIdx.x;
    const unsigned v = (i < n) ? in[i] : 0u;
    tile[threadIdx.x] = v;
    __syncthreads();
    for (int off = 1; off < 1024; off <<= 1) {
      unsigned t = (threadIdx.x >= (unsigned)off) ? tile[threadIdx.x - off] : 0u;
      __syncthreads();
      tile[threadIdx.x] += t;
      __syncthreads();
    }
    if (i < n) out[i] = carry + tile[threadIdx.x] - v;
    __syncthreads();
    if (threadIdx.x == 1023) carry += tile[1023];
    __syncthreads();
  }
}

// --------------------------- node count / compact --------------------------
__global__ __launch_bounds__(256) void node_count_kernel(
    const float* __restrict__ scores, int N,
    const unsigned* __restrict__ state, unsigned* __restrict__ blockCounts) {
  __shared__ unsigned cnt;
  if (threadIdx.x == 0) cnt = 0u;
  __syncthreads();
  const unsigned T = state[2];
  const int i = blockIdx.x * 256 + threadIdx.x;
  if (i < N && fkey(scores[i]) >= T) atomicAdd(&cnt, 1u);
  __syncthreads();
  if (threadIdx.x == 0) blockCounts[blockIdx.x] = cnt;
}

__global__ __launch_bounds__(256) void node_compact_kernel(
    const float* __restrict__ scores, int N, int nKeep,
    const unsigned* __restrict__ state, const unsigned* __restrict__ blockOffs,
    int* __restrict__ poolIdx, int* __restrict__ nodeMask,
    int* __restrict__ mapping, float* __restrict__ outPool) {
  __shared__ unsigned sc[256];
  const unsigned T = state[2];
  const int i = blockIdx.x * 256 + threadIdx.x;
  unsigned keep = 0u;
  if (i < N) keep = (fkey(scores[i]) >= T) ? 1u : 0u;
  sc[threadIdx.x] = keep;
  __syncthreads();
  for (int off = 1; off < 256; off <<= 1) {
    unsigned t = (threadIdx.x >= (unsigned)off) ? sc[threadIdx.x - off] : 0u;
    __syncthreads();
    sc[threadIdx.x] += t;
    __syncthreads();
  }
  if (i < N) {
    const int rank = (int)(blockOffs[blockIdx.x] + sc[threadIdx.x] - keep);
    const int kept = (keep && rank < nKeep) ? 1 : 0;   // stable: first nKeep
    nodeMask[i] = kept;
    mapping[i]  = kept ? rank : -1;
    if (kept) { poolIdx[rank] = i; outPool[rank] = (float)i; }
  }
}

// ------------------------------- new_x gather ------------------------------
__global__ __launch_bounds__(256) void gather_x_kernel(
    const float* __restrict__ x, const int* __restrict__ poolIdx,
    float* __restrict__ outX, int nKeep, int F4) {
  const int tid = blockIdx.x * 256 + threadIdx.x;
  const int row = tid / F4;
  const int c4  = tid % F4;
  if (row < nKeep) {
    const int src = poolIdx[row];
    const float4* s = (const float4*)(x + (size_t)src * (size_t)(F4 * 4));
    float4* d = (float4*)(outX + (size_t)row * (size_t)(F4 * 4));
    d[c4] = s[c4];
  }
}

// --------------------------- edge count / compact --------------------------
__global__ __launch_bounds__(256) void edge_count_kernel(
    const int* __restrict__ ei, int E, const int* __restrict__ nodeMask,
    unsigned* __restrict__ blockCounts, float* __restrict__ outPoolEdges) {
  __shared__ unsigned cnt;
  if (threadIdx.x == 0) cnt = 0u;
  __syncthreads();
  const int e = blockIdx.x * 256 + threadIdx.x;
  if (e < E) {
    const int s = ei[e];
    const int d = ei[E + e];
    const int keep = nodeMask[s] & nodeMask[d];
    outPoolEdges[e] = (float)keep;
    if (keep) atomicAdd(&cnt, 1u);
  }
  __syncthreads();
  if (threadIdx.x == 0) blockCounts[blockIdx.x] = cnt;
}

__global__ __launch_bounds__(256) void edge_compact_kernel(
    const int* __restrict__ ei, int E, const float* __restrict__ eattr,
    const int* __restrict__ nodeMask, const int* __restrict__ mapping,
    const unsigned* __restrict__ blockOffs,
    float* __restrict__ outNEI, float* __restrict__ outOrd,
    float* __restrict__ outAttr, int eKept, int Fe) {
  __shared__ unsigned sc[256];
  const int e = blockIdx.x * 256 + threadIdx.x;
  int s = 0, d = 0;
  unsigned keep = 0u;
  if (e < E) {
    s = ei[e];
    d = ei[E + e];
    keep = (unsigned)(nodeMask[s] & nodeMask[d]);
    if (keep) __builtin_prefetch(eattr + (size_t)e * Fe, 0, 1); // global_prefetch_b8
  }
  sc[threadIdx.x] = keep;
  __syncthreads();
  for (int off = 1; off < 256; off <<= 1) {
    unsigned t = (threadIdx.x >= (unsigned)off) ? sc[threadIdx.x - off] : 0u;
    __syncthreads();
    sc[threadIdx.x] += t;
    __syncthreads();
  }
  if (e < E && keep) {
    const int r = (int)(blockOffs[blockIdx.x] + sc[threadIdx.x] - 1u);
    if (r < eKept) {
      outNEI[r]         = (float)s;
      outNEI[eKept + r] = (float)d;
      outOrd[r]         = (float)mapping[s];
      outOrd[eKept + r] = (float)mapping[d];
      const float4* srcA = (const float4*)(eattr + (size_t)e * Fe);
      float4* dstA = (float4*)(outAttr + (size_t)r * Fe);
      const int q4 = Fe >> 2;
      for (int q = 0; q < q4; ++q) dstA[q] = srcA[q];
    }
  }
}

// ------------------------------ host launcher ------------------------------
extern "C" void kernel_launch(void* const* d_in, const int* in_sizes, int n_in,
                              void* d_out, int out_size, void* d_ws, size_t ws_size,
                              hipStream_t stream) {
  const float* x     = (const float*)d_in[0];   // (1,N,F)
  const float* eattr = (const float*)d_in[1];   // (1,E,Fe)
  const float* pw    = (const float*)d_in[2];   // (1,F)
  const float* pb    = (const float*)d_in[3];   // (1,)
  const int*   ei    = (const int*)d_in[4];     // (2,E)

  const int F  = in_sizes[2];
  const int N  = in_sizes[0] / F;
  const int E  = in_sizes[4] / 2;
  const int Fe = in_sizes[1] / E;

  const int nKeep = N / 2;                      // int(N * K_FRAC), K_FRAC=0.5
  const unsigned kSel = (unsigned)(N - N / 2);  // kth = int((1-K_FRAC)*N), 1-based

  // Recover data-dependent edge count from the fixed output size:
  // out = nKeep*F + 2*Ek + 2*Ek + Fe*Ek + nKeep + E
  long long ekll = ((long long)out_size - (long long)nKeep * F - nKeep - E)
                   / (long long)(4 + Fe);
  const int eKept = (ekll > 0) ? (int)ekll : 0;

  float* outX    = (float*)d_out;
  float* outNEI  = outX + (size_t)nKeep * F;
  float* outOrd  = outNEI + (size_t)2 * eKept;
  float* outAttr = outOrd + (size_t)2 * eKept;
  float* outPool = outAttr + (size_t)Fe * eKept;
  float* outPE   = outPool + (size_t)nKeep;

  // Workspace carve-out (256B aligned slices).
  char* ws = (char*)d_ws;
  size_t off = 0;
  auto carve = [&](size_t bytes) -> char* {
    char* p = ws + off;
    off = (off + bytes + 255) & ~(size_t)255;
    return p;
  };
  float*    scores   = (float*)carve((size_t)N * 4);
  int*      nodeMask = (int*)carve((size_t)N * 4);
  int*      mapping  = (int*)carve((size_t)N * 4);
  int*      poolIdx  = (int*)carve((size_t)nKeep * 4);
  const int nbN = (N + 255) / 256;
  const int nbE = (E + 255) / 256;
  unsigned* bcN   = (unsigned*)carve((size_t)nbN * 4);
  unsigned* boN   = (unsigned*)carve((size_t)nbN * 4);
  unsigned* bcE   = (unsigned*)carve((size_t)nbE * 4);
  unsigned* boE   = (unsigned*)carve((size_t)nbE * 4);
  unsigned* hist  = (unsigned*)carve(256 * 4);
  unsigned* state = (unsigned*)carve(4 * 4);
  (void)ws_size; (void)n_in;

  // 1) scores via WMMA GEMV (128 rows per 256-thread block = 8 waves x 16 rows)
  const int scoreBlocks = (N + 127) / 128;
  score_kernel<<<scoreBlocks, 256, 0, stream>>>(x, pw, pb, scores, N, F);

  // 2) radix-select the kSel-th smallest score key
  radix_init_kernel<<<1, 256, 0, stream>>>(state, hist, kSel);
  for (int shift = 24; shift >= 0; shift -= 8) {
    radix_hist_kernel<<<nbN, 256, 0, stream>>>(scores, N, state, hist, shift);
    radix_pick_kernel<<<1, 256, 0, stream>>>(state, hist, shift);
  }

  // 3) stable node compaction (first nKeep indices with score >= threshold)
  node_count_kernel<<<nbN, 256, 0, stream>>>(scores, N, state, bcN);
  scan_exclusive_kernel<<<1, 1024, 0, stream>>>(bcN, boN, nbN);
  node_compact_kernel<<<nbN, 256, 0, stream>>>(scores, N, nKeep, state, boN,
                                               poolIdx, nodeMask, mapping, outPool);

  // 4) gather new_x
  const int F4 = F / 4;
  const int gblocks = ((long long)nKeep * F4 + 255) / 256;
  gather_x_kernel<<<gblocks, 256, 0, stream>>>(x, poolIdx, outX, nKeep, F4);

  // 5) edge filter + stable compaction
  edge_count_kernel<<<nbE, 256, 0, stream>>>(ei, E, nodeMask, bcE, outPE);
  scan_exclusive_kernel<<<1, 1024, 0, stream>>>(bcE, boE, nbE);
  edge_compact_kernel<<<nbE, 256, 0, stream>>>(ei, E, eattr, nodeMask, mapping,
                                               boE, outNEI, outOrd, outAttr,
                                               eKept, Fe);
}